// GroupLinear_5050881540801
// MI455X (gfx1250) — compile-verified
//
#include <hip/hip_runtime.h>

// Grouped GEMM for MI455X (gfx1250, wave32): exact-fp32 WMMA compute core,
// operand staging via Tensor Data Mover (tensor_load_to_lds + TENSORcnt).
// out[t, :] = x[t, :] @ W[g(t)].T   with contiguous row ranges per group.

#define G_    8
#define T_    4096
#define DIN   1024
#define DOUT  4096
#define TM    128
#define TN    128
#define TK    32
#define LDA   36                 // TDM pad: 4-dword pad after every 32 dwords -> 36-float row stride
#define NT    (DOUT / TN)        // 32 n-tiles
#define NK    (DIN / TK)         // 32 k-chunks

typedef __attribute__((ext_vector_type(2))) float v2f;
typedef __attribute__((ext_vector_type(8))) float v8f;
typedef __attribute__((ext_vector_type(4))) unsigned int u32x4;
typedef __attribute__((ext_vector_type(8))) int i32x8;
typedef __attribute__((ext_vector_type(4))) int i32x4;

// LDS byte offset of a __shared__ object (generic -> addrspace(3) -> 32-bit offset)
__device__ __forceinline__ unsigned int lds_byte_addr(const void* p) {
    return (unsigned int)(__SIZE_TYPE__)(const __attribute__((address_space(3))) char*)p;
}

// Issue one TDM 2-D tile load: tile0 x tile1 elements of 4B, row stride stride0,
// zero-fill past (dim0, dim1), LDS padding 4 dwords per 32 dwords (row stride LDA).
__device__ __forceinline__ void tdm_load_2d(unsigned int lds_addr, const void* gaddr,
                                            unsigned int dim0, unsigned int dim1,
                                            unsigned int tile0, unsigned int tile1,
                                            unsigned int stride0) {
    const unsigned long long ga = (unsigned long long)(__SIZE_TYPE__)gaddr;

    u32x4 g0;
    g0[0] = 1u;                                                   // count=1, user D#
    g0[1] = lds_addr;                                             // LDS byte address
    g0[2] = (unsigned int)(ga & 0xFFFFFFFFu);                     // global_addr[31:0]
    g0[3] = (unsigned int)((ga >> 32) & 0x1FFFFFFu) | (2u << 30); // global_addr[56:32] | type=2

    i32x8 g1;
    g1[0] = (int)((2u << 16)    // data_size = 4B
                | (1u << 20)    // pad_enable
                | (4u << 22)    // pad_interval: 32 dwords
                | (3u << 25));  // pad_amount: 4 dwords
    g1[1] = (int)((dim0 & 0xFFFFu) << 16);                              // tensor_dim0[15:0]
    g1[2] = (int)(((dim0 >> 16) & 0xFFFFu) | ((dim1 & 0xFFFFu) << 16)); // dim0[31:16] | dim1[15:0]
    g1[3] = (int)(((dim1 >> 16) & 0xFFFFu) | ((tile0 & 0xFFFFu) << 16));// dim1[31:16] | tile_dim0
    g1[4] = (int)(tile1 & 0xFFFFu);                                     // tile_dim1, tile_dim2=0
    g1[5] = (int)stride0;                                               // tensor_dim0_stride[31:0]
    g1[6] = 0;
    g1[7] = 0;

    const i32x4 z4 = {0, 0, 0, 0};                   // 2-D tile: groups 2/3 unused
    const i32x8 z8 = {0, 0, 0, 0, 0, 0, 0, 0};
    __builtin_amdgcn_tensor_load_to_lds(g0, g1, z4, z4, z8, 0);
}

__global__ __launch_bounds__(256)
void grouped_gemm_wmma_f32(const float* __restrict__ x,
                           const int* __restrict__ offs,
                           const float* __restrict__ w,
                           float* __restrict__ out)
{
    // ---- map blockIdx.x -> (group g, m_tile, n_tile) using device-side offsets ----
    const int b = blockIdx.x;
    int g = -1, local = 0, rowS = 0, rowE = 0;
    int base = 0;
    #pragma unroll
    for (int gg = 0; gg < G_; ++gg) {
        const int s = gg ? offs[gg - 1] : 0;
        const int e = offs[gg];
        const int mt = (e - s + TM - 1) / TM;
        const int cnt = mt * NT;
        if (g < 0 && b < base + cnt) { g = gg; local = b - base; rowS = s; rowE = e; }
        base += cnt;
    }
    if (g < 0) return;   // uniform per block

    const int mTile = local / NT;
    const int nTile = local % NT;
    const int row0  = rowS + mTile * TM;
    const int n0    = nTile * TN;

    __shared__ float lsA[2][TM * LDA];   // 36 KB
    __shared__ float lsW[2][TN * LDA];   // 36 KB

    const float* __restrict__ wg = w + (size_t)g * DOUT * DIN;

    // ---- WMMA fragment indexing (wave32) ----
    const int tid  = threadIdx.x;
    const int lane = tid & 31;
    const int wid  = tid >> 5;           // 8 waves
    const int wm   = (wid & 1) << 6;     // 0 / 64   (M offset of wave)
    const int wn   = (wid >> 1) << 5;    // 0/32/64/96 (N offset of wave)
    const int lm   = lane & 15;          // row (A) / col (B) within frag
    const int lk   = (lane >> 4) << 1;   // K sub-offset: lanes 0-15 -> k+0/1, 16-31 -> k+2/3

    v8f acc[4][2] = {};                  // 64 VGPRs of f32 accumulators

    // issue the two tile DMAs (A and W) for one K-chunk into LDS buffer `buf`
    auto tdm_pair = [&](int k0, int buf) {
        tdm_load_2d(lds_byte_addr(&lsA[buf][0]),
                    x + (size_t)row0 * DIN + k0,
                    (unsigned)(DIN - k0), (unsigned)(T_ - row0),  // zero-fill past end of x
                    TK, TM, DIN);
        tdm_load_2d(lds_byte_addr(&lsW[buf][0]),
                    wg + (size_t)n0 * DIN + k0,
                    (unsigned)(DIN - k0), (unsigned)(DOUT - n0),
                    TK, TN, DIN);
    };

    auto compute = [&](int buf) {
        #pragma unroll
        for (int kk = 0; kk < TK / 4; ++kk) {           // 8 micro-K steps of 4
            v2f af[4], bf[2];
            #pragma unroll
            for (int fm = 0; fm < 4; ++fm)
                af[fm] = *(const v2f*)&lsA[buf][(wm + fm * 16 + lm) * LDA + kk * 4 + lk];
            #pragma unroll
            for (int fn = 0; fn < 2; ++fn)
                bf[fn] = *(const v2f*)&lsW[buf][(wn + fn * 16 + lm) * LDA + kk * 4 + lk];
            #pragma unroll
            for (int fm = 0; fm < 4; ++fm)
                #pragma unroll
                for (int fn = 0; fn < 2; ++fn)
                    acc[fm][fn] = __builtin_amdgcn_wmma_f32_16x16x4_f32(
                        false, af[fm], false, bf[fn],
                        (short)0, acc[fm][fn], false, false);
        }
    };

    // ---- TDM-pipelined K loop: wave 0 drives the DMA, all 8 waves compute ----
    const bool tdmWave = (wid == 0);
    if (tdmWave) {
        tdm_pair(0, 0);
        tdm_pair(TK, 1);
        __builtin_amdgcn_s_wait_tensorcnt(2);   // chunk 0 resident (TDM in-order per wave)
    }
    __syncthreads();

    for (int kc = 0; kc < NK; ++kc) {
        const int cur = kc & 1;
        compute(cur);
        __syncthreads();                         // all waves done reading buf `cur`
        if (tdmWave) {
            if (kc + 2 < NK) {
                tdm_pair((kc + 2) * TK, cur);    // refill just-freed buffer
                __builtin_amdgcn_s_wait_tensorcnt(2);   // chunk kc+1 resident
            } else {
                __builtin_amdgcn_s_wait_tensorcnt(0);
            }
        }
        __syncthreads();                         // publish chunk kc+1 to all waves
    }

    // ---- store C frags (v8f layout: lanes 0-15 M=r, lanes 16-31 M=r+8), mask rows past group end ----
    #pragma unroll
    for (int fm = 0; fm < 4; ++fm) {
        #pragma unroll
        for (int fn = 0; fn < 2; ++fn) {
            const int col = n0 + wn + fn * 16 + lm;
            const int mB  = row0 + wm + fm * 16 + ((lane >> 4) << 3);
            #pragma unroll
            for (int r = 0; r < 8; ++r) {
                const int m = mB + r;
                if (m < rowE) out[(size_t)m * DOUT + col] = acc[fm][fn][r];
            }
        }
    }
}

extern "C" void kernel_launch(void* const* d_in, const int* in_sizes, int n_in,
                              void* d_out, int out_size, void* d_ws, size_t ws_size,
                              hipStream_t stream) {
    (void)in_sizes; (void)n_in; (void)out_size; (void)d_ws; (void)ws_size;
    const float* x    = (const float*)d_in[0];
    const int*   offs = (const int*)d_in[1];
    const float* w    = (const float*)d_in[2];
    float*       out  = (float*)d_out;

    // worst case tiles: (T/TM + G) m-tiles * NT n-tiles; surplus blocks exit early
    const int grid = (T_ / TM + G_) * NT;   // 40 * 32 = 1280
    grouped_gemm_wmma_f32<<<grid, 256, 0, stream>>>(x, offs, w, out);
}